// Model_21260088115737
// MI455X (gfx1250) — compile-verified
//
#include <hip/hip_runtime.h>
#include <cstdint>

// Problem constants (match reference)
#define NUM_TOKENS 8192
#define NUM_SLOTS  16384          // NUM_BLOCKS * BLOCK_SIZE
#define HD         4096           // NUM_HEADS * HEAD_DIM (floats per slot per cache)
#define SLOT_BYTES (HD * 4)       // 16 KB per slot per cache
#define HALF_ELEMS ((size_t)NUM_SLOTS * (size_t)HD)   // floats per cache

// ---------------------------------------------------------------------------
// Kernel 1: init inverse map (slot -> token) to -1. 64 KB in d_ws, redone
// every call so the launch is deterministic / graph-replay safe.
// ---------------------------------------------------------------------------
__global__ void inv_init(int* __restrict__ inv) {
    int i = blockIdx.x * blockDim.x + threadIdx.x;
    if (i < NUM_SLOTS) inv[i] = -1;
}

// ---------------------------------------------------------------------------
// Kernel 2: inv[slot_mapping[t]] = t. slot_mapping entries are unique
// (permutation subset), so plain stores are race-free.
// ---------------------------------------------------------------------------
__global__ void inv_fill(const int* __restrict__ slot_mapping, int* __restrict__ inv) {
    int t = blockIdx.x * blockDim.x + threadIdx.x;
    if (t < NUM_TOKENS) inv[slot_mapping[t]] = t;
}

// ---------------------------------------------------------------------------
// Kernel 3: one workgroup per slot. Source = token K/V if this slot is being
// overwritten, else the old cache. Data is streamed through LDS with the
// gfx1250 async global<->LDS DMA path (ASYNCcnt-tracked), 16 B per lane per
// issue: 256 lanes x 16 B x 4 issues = 16 KB per cache per block.
// ---------------------------------------------------------------------------
__global__ __launch_bounds__(256) void gather_write(
    const float* __restrict__ key,
    const float* __restrict__ key_cache,
    const float* __restrict__ value,
    const float* __restrict__ value_cache,
    const int*   __restrict__ inv,
    float* __restrict__ out_k,
    float* __restrict__ out_v) {

    __shared__ unsigned char smem[2 * SLOT_BYTES];   // 32 KB: K tile + V tile

    const int slot = blockIdx.x;
    const int tid  = threadIdx.x;
    const int t    = inv[slot];                      // uniform across block

    // Uniform source select: token data (overwritten slot) or old cache.
    const char* srcK = (const char*)((t >= 0) ? (key   + (size_t)t    * HD)
                                              : (key_cache   + (size_t)slot * HD));
    const char* srcV = (const char*)((t >= 0) ? (value + (size_t)t    * HD)
                                              : (value_cache + (size_t)slot * HD));
    char* dstK = (char*)(out_k + (size_t)slot * HD);
    char* dstV = (char*)(out_v + (size_t)slot * HD);

    // LDS byte offsets (workgroup-relative; low 32 bits of generic pointer).
    const uint32_t lbase = (uint32_t)(size_t)(void*)smem;
    const uint32_t lk = lbase + (uint32_t)tid * 16u;
    const uint32_t lv = lbase + (uint32_t)SLOT_BYTES + (uint32_t)tid * 16u;

    // ---- async loads: global -> LDS (b128 per lane) ----
    #pragma unroll
    for (int i = 0; i < 4; ++i) {
        uint64_t ga = (uint64_t)(srcK + tid * 16 + i * 4096);
        uint32_t la = lk + (uint32_t)(i * 4096);
        asm volatile("global_load_async_to_lds_b128 %0, %1, off"
                     :: "v"(la), "v"(ga) : "memory");
    }
    #pragma unroll
    for (int i = 0; i < 4; ++i) {
        uint64_t ga = (uint64_t)(srcV + tid * 16 + i * 4096);
        uint32_t la = lv + (uint32_t)(i * 4096);
        asm volatile("global_load_async_to_lds_b128 %0, %1, off"
                     :: "v"(la), "v"(ga) : "memory");
    }

    // Wait for all async loads to land in LDS before streaming back out.
    asm volatile("s_wait_asynccnt 0" ::: "memory");

    // ---- async stores: LDS -> global (b128 per lane) ----
    #pragma unroll
    for (int i = 0; i < 4; ++i) {
        uint64_t ga = (uint64_t)(dstK + tid * 16 + i * 4096);
        uint32_t la = lk + (uint32_t)(i * 4096);
        asm volatile("global_store_async_from_lds_b128 %0, %1, off"
                     :: "v"(ga), "v"(la) : "memory");
    }
    #pragma unroll
    for (int i = 0; i < 4; ++i) {
        uint64_t ga = (uint64_t)(dstV + tid * 16 + i * 4096);
        uint32_t la = lv + (uint32_t)(i * 4096);
        asm volatile("global_store_async_from_lds_b128 %0, %1, off"
                     :: "v"(ga), "v"(la) : "memory");
    }

    // Drain async stores (S_ENDPGM also implies wait-idle; explicit for clarity).
    asm volatile("s_wait_asynccnt 0" ::: "memory");
}

// ---------------------------------------------------------------------------
// Launch: inputs in setup_inputs() order:
//   d_in[0]=key  d_in[1]=key_cache  d_in[2]=slot_mapping  d_in[3]=value
//   d_in[4]=value_cache
// Output: [key_cache_out | value_cache_out] concatenated flat.
// ---------------------------------------------------------------------------
extern "C" void kernel_launch(void* const* d_in, const int* in_sizes, int n_in,
                              void* d_out, int out_size, void* d_ws, size_t ws_size,
                              hipStream_t stream) {
    const float* key          = (const float*)d_in[0];
    const float* key_cache    = (const float*)d_in[1];
    const int*   slot_mapping = (const int*)d_in[2];   // jnp int64 -> int32 (x64 off)
    const float* value        = (const float*)d_in[3];
    const float* value_cache  = (const float*)d_in[4];

    float* out_k = (float*)d_out;
    float* out_v = out_k + HALF_ELEMS;

    int* inv = (int*)d_ws;   // 64 KB scratch for slot -> token inverse map

    inv_init<<<NUM_SLOTS / 256, 256, 0, stream>>>(inv);
    inv_fill<<<NUM_TOKENS / 256, 256, 0, stream>>>(slot_mapping, inv);
    gather_write<<<NUM_SLOTS, 256, 0, stream>>>(key, key_cache, value, value_cache,
                                                inv, out_k, out_v);
}